// SimEncoder_60000693125366
// MI455X (gfx1250) — compile-verified
//
#include <hip/hip_runtime.h>
#include <math.h>

// ---------------------------------------------------------------------------
// SimEncoder loss for MI455X (gfx1250, wave32, WMMA).
//
// Pipeline (all on `stream`, workspace in d_ws):
//   deg/inv_deg -> moments(x)->xd[N,384] -> emb=x@W_pre (WMMA f32 16x16x4)
//   -> moments(emb)->hd[N,192] -> h=hd@W_post (WMMA)
//   -> column means, gather+center (K-major), row norms
//   -> fused tile kernel: 16x16 tiles of BOTH sim matrices via WMMA,
//      squared-diff reduced to a scalar (B*B matrices never materialized)
//   -> loss = sqrt(acc / B^2)
// ---------------------------------------------------------------------------

typedef __attribute__((ext_vector_type(2))) float v2f;
typedef __attribute__((ext_vector_type(8))) float v8f;

#define EPS_ZERO 1e-16f
#define EPS_NORM 1e-8f

#if __has_builtin(__builtin_amdgcn_wmma_f32_16x16x4_f32)
__device__ __forceinline__ v8f wmma4(v2f a, v2f b, v8f c) {
  // D = A(16x4 f32) * B(4x16 f32) + C(16x16 f32)
  return __builtin_amdgcn_wmma_f32_16x16x4_f32(
      /*neg_a=*/false, a, /*neg_b=*/false, b,
      /*c_mod=*/(short)0, c, /*reuse_a=*/false, /*reuse_b=*/false);
}
#else
#warning "wmma f32 16x16x4 builtin not found - scalar fallback (wmma=0)"
__device__ __forceinline__ v8f wmma4(v2f a, v2f b, v8f c) {
#pragma unroll
  for (int j = 0; j < 8; ++j) c[j] += a.x * b.x + a.y * b.y;
  return c;
}
#endif

// --------------------------- utility kernels -------------------------------

__global__ void k_zero(float* __restrict__ p, int n) {
  int i = blockIdx.x * blockDim.x + threadIdx.x;
  if (i < n) p[i] = 0.0f;
}

__global__ void k_deg(const int* __restrict__ row, float* __restrict__ deg, int E) {
  int e = blockIdx.x * blockDim.x + threadIdx.x;
  if (e < E) atomicAdd(&deg[row[e]], 1.0f);
}

__global__ void k_invdeg(const float* __restrict__ deg, float* __restrict__ inv, int N) {
  int i = blockIdx.x * blockDim.x + threadIdx.x;
  if (i < N) {
    float d = deg[i];
    inv[i] = (d > 0.0f) ? 1.0f / fmaxf(d, 1.0f) : 0.0f;
  }
}

// out[row[e], off+f] += V[col[e], f]    (mean-agg numerator)
__global__ void k_scatter_sum(const float* __restrict__ V, int ldv,
                              const int* __restrict__ row, const int* __restrict__ col,
                              float* __restrict__ acc, int ldo, int off,
                              int F, int total) {
  int gid = blockIdx.x * blockDim.x + threadIdx.x;
  if (gid >= total) return;
  int e = gid / F, f = gid - e * F;
  atomicAdd(&acc[(size_t)row[e] * ldo + off + f], V[(size_t)col[e] * ldv + f]);
}

// acc[i, f] *= inv_deg[i]   (finalize mu, stored at column offset 0)
__global__ void k_scale_mu(float* __restrict__ acc, const float* __restrict__ inv_deg,
                           int ldo, int F, int total) {
  int gid = blockIdx.x * blockDim.x + threadIdx.x;
  if (gid >= total) return;
  int i = gid / F, f = gid - i * F;
  acc[(size_t)i * ldo + f] *= inv_deg[i];
}

// sigma/gamma numerators: acc[r,F+f] += (v[c,f]-mu[c,f])^2 ; acc[r,2F+f] += v^3
__global__ void k_scatter_m2(const float* __restrict__ V, int ldv,
                             const int* __restrict__ row, const int* __restrict__ col,
                             float* __restrict__ acc, int ldo, int F, int total) {
  int gid = blockIdx.x * blockDim.x + threadIdx.x;
  if (gid >= total) return;
  int e = gid / F, f = gid - e * F;
  int r = row[e], c = col[e];
  float xv = V[(size_t)c * ldv + f];
  float mu = acc[(size_t)c * ldo + f];  // mu already finalized at offset 0
  float t = (xv - mu) * (xv - mu);
  atomicAdd(&acc[(size_t)r * ldo + F + f], t);
  atomicAdd(&acc[(size_t)r * ldo + 2 * F + f], xv * xv * xv);
}

// sigma = sqrt(where(s==0,eps,s)); gamma = sign(g)*|g|^(1/3)  (g==0 -> eps)
__global__ void k_fin_m(float* __restrict__ acc, const float* __restrict__ inv_deg,
                        int ldo, int F, int total) {
  int gid = blockIdx.x * blockDim.x + threadIdx.x;
  if (gid >= total) return;
  int i = gid / F, f = gid - i * F;
  float id = inv_deg[i];
  size_t base = (size_t)i * ldo;
  float sg = acc[base + F + f] * id;
  sg = (sg == 0.0f) ? EPS_ZERO : sg;
  acc[base + F + f] = sqrtf(sg);
  float g = acc[base + 2 * F + f] * id;
  g = (g == 0.0f) ? EPS_ZERO : g;
  acc[base + 2 * F + f] = copysignf(cbrtf(fabsf(g)), g);
}

// column sums over N rows (row-major H[N,F]); coalesced, atomic per block-chunk
__global__ void k_colsum(const float* __restrict__ H, int N, int F,
                         float* __restrict__ colsum) {
  int c = blockIdx.x * blockDim.x + threadIdx.x;
  if (c >= F) return;
  int r0 = blockIdx.y * 512;
  int r1 = min(r0 + 512, N);
  float s = 0.0f;
  for (int r = r0; r < r1; ++r) s += H[(size_t)r * F + c];
  atomicAdd(&colsum[c], s);
}

// outT[k, i] = H[idx[i], k] - colsum[k]/N    (gather + center, K-major output)
__global__ void k_gather_t(const float* __restrict__ H, int ld,
                           const int* __restrict__ idx,
                           const float* __restrict__ colsum, float invN,
                           float* __restrict__ outT, int Bsz) {
  int i = blockIdx.x * blockDim.x + threadIdx.x;
  int k = blockIdx.y;
  if (i >= Bsz) return;
  float v = H[(size_t)idx[i] * ld + k] - colsum[k] * invN;
  outT[(size_t)k * Bsz + i] = v;
}

// nrm[i] = ||column i of K-major T||  (coalesced across i)
__global__ void k_colnorm(const float* __restrict__ T, int F, int Bsz,
                          float* __restrict__ nrm) {
  int i = blockIdx.x * blockDim.x + threadIdx.x;
  if (i >= Bsz) return;
  float s = 0.0f;
  for (int k = 0; k < F; ++k) {
    float v = T[(size_t)k * Bsz + i];
    s += v * v;
  }
  nrm[i] = sqrtf(s);
}

// ------------------------------ WMMA GEMM ----------------------------------
// C[M,ldc] = A[M,lda(row-major)] * B[K,ldb(row-major)] + bias, one wave per
// 16x16 tile, K looped in steps of 4 via V_WMMA_F32_16X16X4_F32.
// A frag: lane(l16,hi) holds A[m0+l16, k0+2*hi .. +1]  (ISA 16x4 f32 layout)
// B frag: lane(l16,hi) holds B[k0+2*hi .. +1, n0+l16]
__global__ __launch_bounds__(32) void k_gemm16(
    const float* __restrict__ A, int lda,
    const float* __restrict__ B, int ldb,
    const float* __restrict__ bias,
    float* __restrict__ C, int ldc, int K) {
  const int lane = threadIdx.x;
  const int hi = lane >> 4, l16 = lane & 15;
  const int m0 = blockIdx.x * 16, n0 = blockIdx.y * 16;
  v8f c = {0.f, 0.f, 0.f, 0.f, 0.f, 0.f, 0.f, 0.f};
  const float* arow = A + (size_t)(m0 + l16) * lda;
  for (int k0 = 0; k0 < K; k0 += 4) {
    int kb = k0 + 2 * hi;
    v2f a;
    a.x = arow[kb];
    a.y = arow[kb + 1];
    v2f b;
    b.x = B[(size_t)kb * ldb + n0 + l16];
    b.y = B[(size_t)(kb + 1) * ldb + n0 + l16];
    c = wmma4(a, b, c);
  }
  float bv = bias[n0 + l16];
#pragma unroll
  for (int j = 0; j < 8; ++j) {  // C/D layout: lane L, vgpr j -> (j+8*hi, l16)
    int m = m0 + j + 8 * hi;
    C[(size_t)m * ldc + n0 + l16] = c[j] + bv;
  }
}

// ------------------------- fused similarity loss ---------------------------
// One wave computes a 16x64 strip of BOTH sim matrices (A-fragment reused over
// 4 B-tiles for 2.5x L2 intensity), divides by norm products, and reduces
// (emb_sim - x_sim)^2 into a scalar. K-major operands -> coalesced frag loads.
__global__ __launch_bounds__(32) void k_sim_loss(
    const float* __restrict__ XsT, const float* __restrict__ XtT,  // 384 x B
    const float* __restrict__ HsT, const float* __restrict__ HtT,  // 64 x B
    const float* __restrict__ nxs, const float* __restrict__ nxt,
    const float* __restrict__ nhs, const float* __restrict__ nht,
    float* __restrict__ acc_out) {
  const int Bsz = 8192;
  const int lane = threadIdx.x, hi = lane >> 4, l16 = lane & 15;
  const int m0 = blockIdx.x * 16;
  const int n0 = blockIdx.y * 64;

  v8f z = {0.f, 0.f, 0.f, 0.f, 0.f, 0.f, 0.f, 0.f};
  v8f cx[4], ch[4];
#pragma unroll
  for (int t = 0; t < 4; ++t) { cx[t] = z; ch[t] = z; }

  for (int k0 = 0; k0 < 384; k0 += 4) {  // x_sim numerator, K = 384
    int kb = k0 + 2 * hi;
    v2f a;
    a.x = XsT[kb * Bsz + m0 + l16];
    a.y = XsT[(kb + 1) * Bsz + m0 + l16];
#pragma unroll
    for (int t = 0; t < 4; ++t) {
      v2f b;
      b.x = XtT[kb * Bsz + n0 + 16 * t + l16];
      b.y = XtT[(kb + 1) * Bsz + n0 + 16 * t + l16];
      cx[t] = wmma4(a, b, cx[t]);
    }
  }
  for (int k0 = 0; k0 < 64; k0 += 4) {  // emb_sim numerator, K = 64
    int kb = k0 + 2 * hi;
    v2f a;
    a.x = HsT[kb * Bsz + m0 + l16];
    a.y = HsT[(kb + 1) * Bsz + m0 + l16];
#pragma unroll
    for (int t = 0; t < 4; ++t) {
      v2f b;
      b.x = HtT[kb * Bsz + n0 + 16 * t + l16];
      b.y = HtT[(kb + 1) * Bsz + n0 + 16 * t + l16];
      ch[t] = wmma4(a, b, ch[t]);
    }
  }

  float s = 0.0f;
#pragma unroll
  for (int t = 0; t < 4; ++t) {
    float nxc = nxt[n0 + 16 * t + l16];
    float nhc = nht[n0 + 16 * t + l16];
#pragma unroll
    for (int j = 0; j < 8; ++j) {
      int m = m0 + j + 8 * hi;
      float dx = cx[t][j] / (nxs[m] * nxc + EPS_NORM);
      float dh = ch[t][j] / (nhs[m] * nhc + EPS_NORM);
      float d = dh - dx;
      s += d * d;
    }
  }
#pragma unroll
  for (int off = 16; off > 0; off >>= 1) s += __shfl_xor(s, off, 32);
  if (lane == 0) atomicAdd(acc_out, s);
}

__global__ void k_fin_loss(const float* __restrict__ acc, float* __restrict__ out) {
  out[0] = sqrtf(acc[0] * (1.0f / (8192.0f * 8192.0f)));
}

// ------------------------------- launcher ----------------------------------

extern "C" void kernel_launch(void* const* d_in, const int* in_sizes, int n_in,
                              void* d_out, int out_size, void* d_ws, size_t ws_size,
                              hipStream_t stream) {
  const int N = 50000, E = 800000, Bsz = 8192, FIN = 128, EMB = 64;
  const int LDX = 3 * FIN;  // 384
  const int LDH = 3 * EMB;  // 192

  const float* x      = (const float*)d_in[0];
  const int*   ei     = (const int*)d_in[1];
  const int*   src    = (const int*)d_in[2];
  const int*   tar    = (const int*)d_in[3];
  const float* W_pre  = (const float*)d_in[4];
  const float* b_pre  = (const float*)d_in[5];
  const float* W_post = (const float*)d_in[6];
  const float* b_post = (const float*)d_in[7];
  const int* row = ei;      // edge_index[0]
  const int* col = ei + E;  // edge_index[1]
  float* out = (float*)d_out;

  // workspace carve-up
  float* W = (float*)d_ws;
  size_t o = 0;
  auto alloc = [&](size_t n) { float* p = W + o; o += n; return p; };
  float* deg     = alloc(N);
  float* inv_deg = alloc(N);
  float* xd      = alloc((size_t)N * LDX);   // [mu | sigma | gamma] of x
  float* hd      = alloc((size_t)N * LDH);   // [mu | sigma | gamma] of emb
  float* emb     = alloc((size_t)N * EMB);
  float* h       = alloc((size_t)N * EMB);
  float* colsum  = alloc(LDX + EMB + 1);     // colsum_x | colsum_h | lossAcc
  float* colsum_h = colsum + LDX;
  float* lacc    = colsum + LDX + EMB;
  float* XsT = alloc((size_t)LDX * Bsz);
  float* XtT = alloc((size_t)LDX * Bsz);
  float* HsT = alloc((size_t)EMB * Bsz);
  float* HtT = alloc((size_t)EMB * Bsz);
  float* nxs = alloc(Bsz);
  float* nxt = alloc(Bsz);
  float* nhs = alloc(Bsz);
  float* nht = alloc(Bsz);
  (void)ws_size; (void)in_sizes; (void)n_in; (void)out_size;

  auto g = [](long n) { return (int)((n + 255) / 256); };

  // zero accumulators (harness does not re-poison between replays)
  k_zero<<<g(N), 256, 0, stream>>>(deg, N);
  k_zero<<<g((long)N * LDX), 256, 0, stream>>>(xd, N * LDX);
  k_zero<<<g((long)N * LDH), 256, 0, stream>>>(hd, N * LDH);
  k_zero<<<g(LDX + EMB + 1), 256, 0, stream>>>(colsum, LDX + EMB + 1);

  // degree / inverse degree
  k_deg<<<g(E), 256, 0, stream>>>(row, deg, E);
  k_invdeg<<<g(N), 256, 0, stream>>>(deg, inv_deg, N);

  // moments of x -> xd [N, 384]
  k_scatter_sum<<<g((long)E * FIN), 256, 0, stream>>>(x, FIN, row, col, xd, LDX, 0, FIN, E * FIN);
  k_scale_mu<<<g((long)N * FIN), 256, 0, stream>>>(xd, inv_deg, LDX, FIN, N * FIN);
  k_scatter_m2<<<g((long)E * FIN), 256, 0, stream>>>(x, FIN, row, col, xd, LDX, FIN, E * FIN);
  k_fin_m<<<g((long)N * FIN), 256, 0, stream>>>(xd, inv_deg, LDX, FIN, N * FIN);

  // emb = x @ W_pre + b_pre    (50000 = 3125*16 exactly -> full waves, EXEC=~0)
  dim3 ge(N / 16, EMB / 16);
  k_gemm16<<<ge, 32, 0, stream>>>(x, FIN, W_pre, EMB, b_pre, emb, EMB, FIN);

  // moments of emb -> hd [N, 192]
  k_scatter_sum<<<g((long)E * EMB), 256, 0, stream>>>(emb, EMB, row, col, hd, LDH, 0, EMB, E * EMB);
  k_scale_mu<<<g((long)N * EMB), 256, 0, stream>>>(hd, inv_deg, LDH, EMB, N * EMB);
  k_scatter_m2<<<g((long)E * EMB), 256, 0, stream>>>(emb, EMB, row, col, hd, LDH, EMB, E * EMB);
  k_fin_m<<<g((long)N * EMB), 256, 0, stream>>>(hd, inv_deg, LDH, EMB, N * EMB);

  // h = hd @ W_post + b_post
  k_gemm16<<<ge, 32, 0, stream>>>(hd, LDH, W_post, EMB, b_post, h, EMB, LDH);

  // global column means (centering constants)
  dim3 gcx((LDX + 63) / 64, (N + 511) / 512);
  k_colsum<<<gcx, 64, 0, stream>>>(xd, N, LDX, colsum);
  dim3 gch((EMB + 63) / 64, (N + 511) / 512);
  k_colsum<<<gch, 64, 0, stream>>>(h, N, EMB, colsum_h);

  // gather + center into K-major operands, then row norms
  const float invN = 1.0f / (float)N;
  dim3 ggx(Bsz / 256, LDX), ggh(Bsz / 256, EMB);
  k_gather_t<<<ggx, 256, 0, stream>>>(xd, LDX, src, colsum, invN, XsT, Bsz);
  k_gather_t<<<ggx, 256, 0, stream>>>(xd, LDX, tar, colsum, invN, XtT, Bsz);
  k_gather_t<<<ggh, 256, 0, stream>>>(h, EMB, src, colsum_h, invN, HsT, Bsz);
  k_gather_t<<<ggh, 256, 0, stream>>>(h, EMB, tar, colsum_h, invN, HtT, Bsz);
  k_colnorm<<<Bsz / 256, 256, 0, stream>>>(XsT, LDX, Bsz, nxs);
  k_colnorm<<<Bsz / 256, 256, 0, stream>>>(XtT, LDX, Bsz, nxt);
  k_colnorm<<<Bsz / 256, 256, 0, stream>>>(HsT, EMB, Bsz, nhs);
  k_colnorm<<<Bsz / 256, 256, 0, stream>>>(HtT, EMB, Bsz, nht);

  // fused sim matmuls + squared-diff reduction (60 GFLOP of WMMA, zero B*B I/O)
  dim3 gs(Bsz / 16, Bsz / 64);
  k_sim_loss<<<gs, 32, 0, stream>>>(XsT, XtT, HsT, HtT, nxs, nxt, nhs, nht, lacc);

  k_fin_loss<<<1, 1, 0, stream>>>(lacc, out);
}